// VQVAEModel_89335319756898
// MI455X (gfx1250) — compile-verified
//
#include <hip/hip_runtime.h>

typedef float v2f __attribute__((ext_vector_type(2)));
typedef float v8f __attribute__((ext_vector_type(8)));

#define N_TOT   131072      // B*H*W
#define D_DIM   64
#define K_CB    512
#define HW      1024
#define Q_OFF   1           // d_out layout: [loss][quantized 8388608][perplexity][onehot 67108864]
#define P_OFF   8388609
#define OH_OFF  8388610

// ---------------------------------------------------------------------------
// Kernel 1: argmin over codebook via V_WMMA_F32_16X16X4_F32.
// Codebook staged to LDS pre-swizzled into B-fragment order:
//   B value for (code-tile t, k-step s, lane, j) = codebook[t*16 + lane%16][4*s + 2*(lane/16) + j]
// stored lane-major so each wave reads consecutive 8B per lane (bank-conflict free).
// A fragment (16x4 f32): lane holds A[M = lane%16][K = 2*(lane/16) + j].
// C/D layout: lane holds rows m = v + 8*(lane/16), column n = lane%16.
// ---------------------------------------------------------------------------
__global__ __launch_bounds__(256) void vq_argmin_kernel(const float* __restrict__ inputs,
                                                        const float* __restrict__ codebook,
                                                        int* __restrict__ idx_out) {
  extern __shared__ float smem[];          // [0,32768): B frags; [32768,33280): ||e||^2
  float* bfrag = smem;
  float* enorm = smem + 32768;
  const int tid = threadIdx.x;

  // Stage codebook (coalesced global read, swizzled LDS write)
  for (int i = tid; i < K_CB * D_DIM; i += 256) {
    int c = i >> 6, d = i & 63;
    int t = c >> 4, n = c & 15;
    int s = d >> 2, kr = d & 3;
    int half = kr >> 1, j = kr & 1;
    int lane = half * 16 + n;
    bfrag[(((t * 16 + s) * 32 + lane) << 1) + j] = codebook[i];
  }
  // Codebook squared norms
  for (int c = tid; c < K_CB; c += 256) {
    float s = 0.f;
    for (int d = 0; d < D_DIM; ++d) { float v = codebook[c * 64 + d]; s += v * v; }
    enorm[c] = s;
  }
  __syncthreads();

  const int lane = tid & 31;
  const int wave = tid >> 5;
  const int tile = blockIdx.x * 8 + wave;       // 1024 blocks * 8 waves = 8192 row tiles
  const int rowBase = tile * 16;                // 16 rows per tile; never crosses a batch (16 | 1024)
  const int b  = rowBase >> 10;
  const int hw = (rowBase & 1023) + (lane & 15);
  const int half = lane >> 4;
  const float* xbase = inputs + (size_t)b * (D_DIM * HW) + hw;

  // A fragments: BCHW layout -> elements d and d+1 are HW apart, two b32 loads each
  v2f a[16];
#pragma unroll
  for (int s = 0; s < 16; ++s) {
    int d = 4 * s + 2 * half;
    a[s].x = xbase[(size_t)d * HW];
    a[s].y = xbase[(size_t)(d + 1) * HW];
  }

  float best[8];
  int   bidx[8];
#pragma unroll
  for (int v = 0; v < 8; ++v) { best[v] = 3.4e38f; bidx[v] = 0; }

  for (int t = 0; t < 32; ++t) {               // 32 code tiles of 16
    v8f acc = {0.f, 0.f, 0.f, 0.f, 0.f, 0.f, 0.f, 0.f};
    const float* bp = bfrag + ((size_t)(t * 16) * 32 + lane) * 2;
#pragma unroll
    for (int s = 0; s < 16; ++s) {             // K = 64 in steps of 4
      v2f bb = *(const v2f*)(bp + s * 64);
      acc = __builtin_amdgcn_wmma_f32_16x16x4_f32(false, a[s], false, bb,
                                                  (short)0, acc, false, false);
    }
    int code = t * 16 + (lane & 15);
    float nv = enorm[code];
#pragma unroll
    for (int v = 0; v < 8; ++v) {              // ||x||^2 constant per row: drop it
      float dist = nv - 2.0f * acc[v];
      if (dist < best[v]) { best[v] = dist; bidx[v] = code; }  // strict < keeps lowest code on ties
    }
  }

  // Min-reduce (with lowest-index tie-break) across the 16 column lanes of each half
#pragma unroll
  for (int off = 1; off < 16; off <<= 1) {
#pragma unroll
    for (int v = 0; v < 8; ++v) {
      float ob = __shfl_xor(best[v], off, 32);
      int   oi = __shfl_xor(bidx[v], off, 32);
      if (ob < best[v] || (ob == best[v] && oi < bidx[v])) { best[v] = ob; bidx[v] = oi; }
    }
  }
  if ((lane & 15) == 0) {
#pragma unroll
    for (int v = 0; v < 8; ++v)
      idx_out[rowBase + half * 8 + v] = bidx[v];   // row m = v + 8*half
  }
}

// ---------------------------------------------------------------------------
// Kernel 2: gather quantized (BCHW, coalesced per d-plane), commitment-loss
// partial sums (deterministic), code-usage histogram (int atomics).
// ---------------------------------------------------------------------------
__global__ __launch_bounds__(256) void vq_quant_loss_kernel(const float* __restrict__ inputs,
                                                            const float* __restrict__ codebook,
                                                            const int* __restrict__ idx,
                                                            float* __restrict__ out,
                                                            unsigned* __restrict__ counts,
                                                            float* __restrict__ partials) {
  __shared__ float sh[256];
  const int n  = blockIdx.x * 256 + threadIdx.x;   // 512 blocks cover N_TOT
  const int b  = n >> 10, hw = n & 1023;
  const int code = idx[n];
  atomicAdd(&counts[code], 1u);
  const float* crow = codebook + code * 64;
  const float* xb = inputs + (size_t)b * (D_DIM * HW) + hw;
  float* qb = out + Q_OFF + (size_t)b * (D_DIM * HW) + hw;
  float acc = 0.f;
#pragma unroll 4
  for (int d = 0; d < D_DIM; ++d) {
    float q = crow[d];
    float x = xb[(size_t)d * HW];
    float st = x + (q - x);                       // straight-through, matches reference rounding
    __builtin_nontemporal_store(st, &qb[(size_t)d * HW]);
    float diff = q - x;
    acc += diff * diff;
  }
  sh[threadIdx.x] = acc;
  __syncthreads();
  for (int s = 128; s > 0; s >>= 1) {
    if (threadIdx.x < s) sh[threadIdx.x] += sh[threadIdx.x + s];
    __syncthreads();
  }
  if (threadIdx.x == 0) partials[blockIdx.x] = sh[0];
}

// ---------------------------------------------------------------------------
// Kernel 3: one-hot fill [B, K, HW] — 268 MB streaming non-temporal write.
// ---------------------------------------------------------------------------
__global__ __launch_bounds__(256) void vq_onehot_kernel(const int* __restrict__ idx,
                                                        float* __restrict__ out) {
  size_t i = (size_t)blockIdx.x * 256 + threadIdx.x;   // float2 index
  size_t e = i * 2;
  int n_hw  = (int)(e & 1023);
  size_t bk = e >> 10;
  int k = (int)(bk & 511);
  int b = (int)(bk >> 9);
  int i0 = idx[b * 1024 + n_hw];
  int i1 = idx[b * 1024 + n_hw + 1];
  v2f val;
  val.x = (i0 == k) ? 1.0f : 0.0f;
  val.y = (i1 == k) ? 1.0f : 0.0f;
  __builtin_nontemporal_store(val, (v2f*)(out + OH_OFF + e));  // 8B aligned (OH_OFF even)
}

// ---------------------------------------------------------------------------
// Kernel 4: finalize loss + perplexity (deterministic fixed-order trees).
// ---------------------------------------------------------------------------
__global__ __launch_bounds__(512) void vq_finalize_kernel(const float* __restrict__ partials,
                                                          const unsigned* __restrict__ counts,
                                                          float* __restrict__ out) {
  __shared__ float sh[512];
  const int t = threadIdx.x;
  sh[t] = partials[t];
  __syncthreads();
  for (int s = 256; s > 0; s >>= 1) { if (t < s) sh[t] += sh[t + s]; __syncthreads(); }
  float loss = 0.25f * sh[0] / 8388608.0f;      // 0.25 * mean((q-x)^2) over B*H*W*D
  __syncthreads();
  float p = (float)counts[t] / 131072.0f;
  sh[t] = p * logf(p + 1e-10f);
  __syncthreads();
  for (int s = 256; s > 0; s >>= 1) { if (t < s) sh[t] += sh[t + s]; __syncthreads(); }
  if (t == 0) { out[0] = loss; out[P_OFF] = expf(-sh[0]); }
}

extern "C" void kernel_launch(void* const* d_in, const int* in_sizes, int n_in,
                              void* d_out, int out_size, void* d_ws, size_t ws_size,
                              hipStream_t stream) {
  const float* inputs   = (const float*)d_in[0];   // [128,64,32,32] f32
  const float* codebook = (const float*)d_in[1];   // [512,64] f32
  float* out = (float*)d_out;

  int*      idx      = (int*)d_ws;                                        // 131072 ints
  unsigned* counts   = (unsigned*)((char*)d_ws + 131072 * sizeof(int));   // 512
  float*    partials = (float*)((char*)d_ws + 131072 * sizeof(int) + 512 * sizeof(unsigned)); // 512

  hipMemsetAsync(counts, 0, 512 * sizeof(unsigned), stream);  // atomics need zeroed counts each call

  // 130 KB dynamic LDS (codebook B-fragments + norms) — WGP has 320 KB
  vq_argmin_kernel<<<1024, 256, (32768 + 512) * sizeof(float), stream>>>(inputs, codebook, idx);
  vq_quant_loss_kernel<<<512, 256, 0, stream>>>(inputs, codebook, idx, out, counts, partials);
  vq_onehot_kernel<<<131072, 256, 0, stream>>>(idx, out);
  vq_finalize_kernel<<<1, 512, 0, stream>>>(partials, counts, out);
}